// NativeHeteroGNN_57827439674001
// MI455X (gfx1250) — compile-verified
//
#include <hip/hip_runtime.h>
#include <math.h>

typedef __attribute__((ext_vector_type(8))) float v8f;
typedef __attribute__((ext_vector_type(2))) float v2f;

__device__ __forceinline__ float sigmoidf_(float x) { return 1.0f / (1.0f + __expf(-x)); }
__device__ __forceinline__ float gelu_exact(float x) {
  return 0.5f * x * (1.0f + erff(x * 0.70710678118654752440f));
}
// monotonic float<->uint encoding for atomic max on floats
__device__ __forceinline__ unsigned enc_f32(float x) {
  unsigned u = __float_as_uint(x);
  return (u & 0x80000000u) ? ~u : (u | 0x80000000u);
}
__device__ __forceinline__ float dec_f32(unsigned u) {
  return (u & 0x80000000u) ? __uint_as_float(u & 0x7FFFFFFFu) : __uint_as_float(~u);
}

#define ACTA_NONE 0
#define ACTA_GELU 1
#define ACTO_NONE 0
#define ACTO_RELU 1
#define EPI_STORE 0
#define EPI_ACCUM 1
#define EPI_SKIPMIX 2

// Generic f32 WMMA GEMM: C = epi(actO(actA(A) @ B + bias))
// A: M x K (row stride lda). B element (k,n) at B[k*ldbr + n*ldbc] (ldbc!=1 -> transposed).
// Block = 256 threads = 8 waves; wave -> one 16-row strip x NT 16-col tiles.
// The B panel (K x NT*16) is staged transposed into LDS once per block, so the inner
// loop reads each B K-pair as a single aligned ds_load_b64. K, NT compile-time.
template <int NT, int K, int ACTA, int ACTO, int EPI>
__global__ void __launch_bounds__(256)
wmma_gemm_f32(const float* __restrict__ A, int lda,
              const float* __restrict__ B, int ldbr, int ldbc,
              const float* __restrict__ bias,
              float* __restrict__ C, int ldc,
              const float* __restrict__ aux,    // EPI_SKIPMIX: residual (same layout as C)
              const float* __restrict__ skipw,  // EPI_SKIPMIX: scalar gate (pre-sigmoid)
              int M) {
  constexpr int BN = NT * 16;          // panel width (32 or 64 -> power of two)
  constexpr int NSHIFT = (BN == 64) ? 6 : 5;
  constexpr int KP = K + 2;            // even row stride: 8B-aligned float2, bank skew
  __shared__ float bsh[BN * KP];

  const int nbase = blockIdx.y * BN;
  // cooperative stage: bsh[n*KP + k] = B[k, nbase+n]
  for (int idx = threadIdx.x; idx < BN * K; idx += 256) {
    const int n = idx & (BN - 1);
    const int k = idx >> NSHIFT;
    bsh[n * KP + k] = B[(size_t)k * ldbr + (size_t)(nbase + n) * ldbc];
  }
  __syncthreads();

  const int lane = threadIdx.x & 31;
  const int wave = threadIdx.x >> 5;
  const int m0 = (blockIdx.x * 8 + wave) * 16;
  if (m0 >= M) return;
  const int row = lane & 15;   // M index within tile for A; N index for B/C
  const int half = lane >> 4;  // K-pair select (A/B); M+8 select (C)

  v8f acc[NT] = {};
  const float* Arow = A + (size_t)(m0 + row) * lda;
  const float* bl = &bsh[row * KP + 2 * half];  // lane's base in LDS B panel

#pragma unroll
  for (int k0 = 0; k0 < K; k0 += 4) {
    float2 a2 = *reinterpret_cast<const float2*>(Arow + k0 + 2 * half);
    float a0 = a2.x, a1 = a2.y;
    if (ACTA == ACTA_GELU) { a0 = gelu_exact(a0); a1 = gelu_exact(a1); }
    v2f av;
    av.x = a0;
    av.y = a1;
#pragma unroll
    for (int t = 0; t < NT; ++t) {
      float2 b2 = *reinterpret_cast<const float2*>(bl + t * (16 * KP) + k0);
      v2f bv;
      bv.x = b2.x;
      bv.y = b2.y;
      acc[t] = __builtin_amdgcn_wmma_f32_16x16x4_f32(
          false, av, false, bv, (short)0, acc[t], false, false);
    }
  }

  float s = 1.0f, oms = 0.0f;
  if (EPI == EPI_SKIPMIX) {
    s = sigmoidf_(skipw[0]);
    oms = 1.0f - s;
  }

#pragma unroll
  for (int t = 0; t < NT; ++t) {
    const int col = nbase + 16 * t + row;
    const float bb = bias ? bias[col] : 0.0f;
#pragma unroll
    for (int r = 0; r < 8; ++r) {
      const int rr = m0 + r + half * 8;  // C layout: VGPR r -> M=r (lanes0-15) / M=r+8
      float v = acc[t][r] + bb;
      if (ACTO == ACTO_RELU) v = fmaxf(v, 0.0f);
      const size_t idx = (size_t)rr * ldc + col;
      if (EPI == EPI_ACCUM) C[idx] += v;
      else if (EPI == EPI_SKIPMIX) C[idx] = s * v + oms * aux[idx];
      else C[idx] = v;
    }
  }
}

// logits[e,h] = (k_src[src[e],h,:] . qt[dst[e],h,:]) * p[h] / sqrt(32); atomicMax per (dst,h)
__global__ void __launch_bounds__(256)
edge_logits_kernel(const int* __restrict__ src, const int* __restrict__ dst,
                   const float* __restrict__ ksrc, int ldk,
                   const float* __restrict__ qt, const float* __restrict__ p,
                   float* __restrict__ logits, unsigned* __restrict__ maxb, int E) {
  int e = blockIdx.x * blockDim.x + threadIdx.x;
  if (e >= E) return;
  const int si = src[e], di = dst[e];
  const float* kv = ksrc + (size_t)si * ldk;
  const float* qv = qt + (size_t)di * 64;
  const float inv_sqrt_d = 0.17677669529663687f;  // 1/sqrt(32)
#pragma unroll
  for (int h = 0; h < 2; ++h) {
    float acc = 0.0f;
#pragma unroll
    for (int d = 0; d < 32; d += 4) {
      float4 kk = *reinterpret_cast<const float4*>(kv + h * 32 + d);
      float4 qq = *reinterpret_cast<const float4*>(qv + h * 32 + d);
      acc += kk.x * qq.x + kk.y * qq.y + kk.z * qq.z + kk.w * qq.w;
    }
    float lg = acc * p[h] * inv_sqrt_d;
    logits[(size_t)e * 2 + h] = lg;
    atomicMax(maxb + (size_t)di * 2 + h, enc_f32(lg));
  }
}

// ev[i,h] = exp(logit - max[dst,h]); atomicAdd into sum[dst,h]; i spans both relations
__global__ void __launch_bounds__(256)
edge_expsum_kernel(const int* __restrict__ dstA, const int* __restrict__ dstB,
                   float* __restrict__ ev, const unsigned* __restrict__ maxb,
                   float* __restrict__ sumb, int E) {
  int i = blockIdx.x * blockDim.x + threadIdx.x;
  if (i >= 2 * E) return;
  const int di = (i < E) ? dstA[i] : dstB[i - E];
#pragma unroll
  for (int h = 0; h < 2; ++h) {
    float m = dec_f32(maxb[(size_t)di * 2 + h]);
    float x = __expf(ev[(size_t)i * 2 + h] - m);
    ev[(size_t)i * 2 + h] = x;
    atomicAdd(sumb + (size_t)di * 2 + h, x);
  }
}

// aggr[dst,h,:] += attn * v_src[src,h,:]   (raw v; Wv applied later as node GEMM)
__global__ void __launch_bounds__(256)
edge_aggregate_kernel(const int* __restrict__ src, const int* __restrict__ dst,
                      const float* __restrict__ vsrc, int ldv,
                      const float* __restrict__ ev, const float* __restrict__ sumb,
                      float* __restrict__ aggr, int E) {
  int e = blockIdx.x * blockDim.x + threadIdx.x;
  if (e >= E) return;
  const int si = src[e], di = dst[e];
  const float* vv = vsrc + (size_t)si * ldv;
  float* outp = aggr + (size_t)di * 64;
#pragma unroll
  for (int h = 0; h < 2; ++h) {
    float attn = ev[(size_t)e * 2 + h] / fmaxf(sumb[(size_t)di * 2 + h], 1e-16f);
#pragma unroll
    for (int d = 0; d < 32; d += 4) {
      float4 v4 = *reinterpret_cast<const float4*>(vv + h * 32 + d);
      atomicAdd(outp + h * 32 + d + 0, attn * v4.x);
      atomicAdd(outp + h * 32 + d + 1, attn * v4.y);
      atomicAdd(outp + h * 32 + d + 2, attn * v4.z);
      atomicAdd(outp + h * 32 + d + 3, attn * v4.w);
    }
  }
}

// out[i] = hp[i,:] . outW + out_b
__global__ void __launch_bounds__(256)
out_kernel(const float* __restrict__ hp, const float* __restrict__ outW,
           const float* __restrict__ outb, float* __restrict__ out, int M) {
  int i = blockIdx.x * blockDim.x + threadIdx.x;
  if (i >= M) return;
  const float* hr = hp + (size_t)i * 64;
  float acc = outb[0];
#pragma unroll
  for (int f = 0; f < 64; f += 4) {
    float4 h4 = *reinterpret_cast<const float4*>(hr + f);
    float4 w4 = *reinterpret_cast<const float4*>(outW + f);
    acc += h4.x * w4.x + h4.y * w4.y + h4.z * w4.z + h4.w * w4.w;
  }
  out[i] = acc;
}

extern "C" void kernel_launch(void* const* d_in, const int* in_sizes, int n_in,
                              void* d_out, int out_size, void* d_ws, size_t ws_size,
                              hipStream_t stream) {
  const float* x_paper  = (const float*)d_in[0];
  const float* x_author = (const float*)d_in[1];
  const int*   w_src    = (const int*)d_in[2];
  const int*   w_dst    = (const int*)d_in[3];
  const int*   c_src    = (const int*)d_in[4];
  const int*   c_dst    = (const int*)d_in[5];
  // d_in[6..7] = rev edges: provably dead (output depends only on paper branch)
  const float* lin_W_p  = (const float*)d_in[8];
  const float* lin_b_p  = (const float*)d_in[9];
  const float* lin_W_a  = (const float*)d_in[10];
  const float* lin_b_a  = (const float*)d_in[11];
  const float* kqv_W_p  = (const float*)d_in[12];
  const float* kqv_b_p  = (const float*)d_in[13];
  const float* kqv_W_a  = (const float*)d_in[14];
  const float* kqv_b_a  = (const float*)d_in[15];
  const float* a_W_p    = (const float*)d_in[16];
  const float* a_b_p    = (const float*)d_in[17];
  const float* skip_p   = (const float*)d_in[20];
  const float* Wk_w     = (const float*)d_in[22];
  const float* Wv_w     = (const float*)d_in[23];
  const float* p_w      = (const float*)d_in[24];
  const float* Wk_c     = (const float*)d_in[25];
  const float* Wv_c     = (const float*)d_in[26];
  const float* p_c      = (const float*)d_in[27];
  const float* out_W    = (const float*)d_in[31];
  const float* out_b    = (const float*)d_in[32];

  const int NP = in_sizes[0] / 128;  // 200000
  const int NA = in_sizes[1] / 128;  // 100000
  const int E  = in_sizes[2];        // 400000

  float* ws = (float*)d_ws;
  size_t o = 0;
  float* kqv_p = ws + o; o += (size_t)NP * 192;
  float* kqv_a = ws + o; o += (size_t)NA * 192;
  float* xp    = ws + o; o += (size_t)NP * 64;
  float* xa    = ws + o; o += (size_t)NA * 64;
  float* qtw   = ws + o; o += (size_t)NP * 64;   // Wk_writes-transformed q_p
  float* qtc   = ws + o; o += (size_t)NP * 64;   // Wk_cites-transformed q_p
  float* evals = ws + o; o += (size_t)2 * E * 2; // logits, overwritten with exp()
  unsigned* maxb = (unsigned*)(ws + o); o += (size_t)NP * 2;
  float* sumb  = ws + o; o += (size_t)NP * 2;
  float* aggw  = ws + o; o += (size_t)NP * 64;   // raw writes-agg; becomes final agg
  float* aggc  = ws + o; o += (size_t)NP * 64;   // raw cites-agg; reused as hp
  (void)ws_size; (void)n_in; (void)out_size;

  hipMemsetAsync(maxb, 0, (size_t)NP * 2 * sizeof(unsigned), stream);  // enc(-inf-ish)
  hipMemsetAsync(sumb, 0, (size_t)NP * 2 * sizeof(float), stream);
  hipMemsetAsync(aggw, 0, (size_t)NP * 64 * sizeof(float), stream);
  hipMemsetAsync(aggc, 0, (size_t)NP * 64 * sizeof(float), stream);

  dim3 blk(256);
  auto gemm_grid = [](int M, int N, int NT) {
    int strips = (M + 15) / 16;
    return dim3((strips + 7) / 8, (N / 16) / NT);
  };

  // 1) x -> relu(x @ linW + b)
  wmma_gemm_f32<4, 128, ACTA_NONE, ACTO_RELU, EPI_STORE>
      <<<gemm_grid(NP, 64, 4), blk, 0, stream>>>(
      x_paper, 128, lin_W_p, 64, 1, lin_b_p, xp, 64, nullptr, nullptr, NP);
  wmma_gemm_f32<4, 128, ACTA_NONE, ACTO_RELU, EPI_STORE>
      <<<gemm_grid(NA, 64, 4), blk, 0, stream>>>(
      x_author, 128, lin_W_a, 64, 1, lin_b_a, xa, 64, nullptr, nullptr, NA);

  // 2) kqv = x @ kqvW + b  (cols: k=0..63, q=64..127, v=128..191); grid.y = 3
  wmma_gemm_f32<4, 64, ACTA_NONE, ACTO_NONE, EPI_STORE>
      <<<gemm_grid(NP, 192, 4), blk, 0, stream>>>(
      xp, 64, kqv_W_p, 192, 1, kqv_b_p, kqv_p, 192, nullptr, nullptr, NP);
  wmma_gemm_f32<4, 64, ACTA_NONE, ACTO_NONE, EPI_STORE>
      <<<gemm_grid(NA, 192, 4), blk, 0, stream>>>(
      xa, 64, kqv_W_a, 192, 1, kqv_b_a, kqv_a, 192, nullptr, nullptr, NA);

  // 3) qt[n,d] = sum_f q_p[n,f] * Wk[h,d,f]  (B transposed via ldbr=1, ldbc=32)
  for (int h = 0; h < 2; ++h) {
    wmma_gemm_f32<2, 32, ACTA_NONE, ACTO_NONE, EPI_STORE>
        <<<gemm_grid(NP, 32, 2), blk, 0, stream>>>(
        kqv_p + 64 + h * 32, 192, Wk_w + h * 1024, 1, 32, nullptr,
        qtw + h * 32, 64, nullptr, nullptr, NP);
    wmma_gemm_f32<2, 32, ACTA_NONE, ACTO_NONE, EPI_STORE>
        <<<gemm_grid(NP, 32, 2), blk, 0, stream>>>(
        kqv_p + 64 + h * 32, 192, Wk_c + h * 1024, 1, 32, nullptr,
        qtc + h * 32, 64, nullptr, nullptr, NP);
  }

  // 4) per-edge logits + segment max (writes: k_a; cites: k_p)
  dim3 egrid((E + 255) / 256);
  edge_logits_kernel<<<egrid, blk, 0, stream>>>(w_src, w_dst, kqv_a, 192, qtw, p_w,
                                                evals, maxb, E);
  edge_logits_kernel<<<egrid, blk, 0, stream>>>(c_src, c_dst, kqv_p, 192, qtc, p_c,
                                                evals + (size_t)E * 2, maxb, E);

  // 5) exp + segment sum
  dim3 e2grid((2 * E + 255) / 256);
  edge_expsum_kernel<<<e2grid, blk, 0, stream>>>(w_dst, c_dst, evals, maxb, sumb, E);

  // 6) attn-weighted raw-v aggregation per relation
  edge_aggregate_kernel<<<egrid, blk, 0, stream>>>(w_src, w_dst, kqv_a + 128, 192,
                                                   evals, sumb, aggw, E);
  edge_aggregate_kernel<<<egrid, blk, 0, stream>>>(c_src, c_dst, kqv_p + 128, 192,
                                                   evals + (size_t)E * 2, sumb, aggc, E);

  // 7) agg = aggw @ Wv_w + aggc @ Wv_c (per head; in-place into aggw, per-strip safe)
  for (int h = 0; h < 2; ++h) {
    wmma_gemm_f32<2, 32, ACTA_NONE, ACTO_NONE, EPI_STORE>
        <<<gemm_grid(NP, 32, 2), blk, 0, stream>>>(
        aggw + h * 32, 64, Wv_w + h * 1024, 32, 1, nullptr,
        aggw + h * 32, 64, nullptr, nullptr, NP);
    wmma_gemm_f32<2, 32, ACTA_NONE, ACTO_NONE, EPI_ACCUM>
        <<<gemm_grid(NP, 32, 2), blk, 0, stream>>>(
        aggc + h * 32, 64, Wv_c + h * 1024, 32, 1, nullptr,
        aggw + h * 32, 64, nullptr, nullptr, NP);
  }

  // 8) hp = sigmoid(skip)*(gelu(agg) @ aW + ab) + (1-sigmoid(skip))*xp  -> aggc (reused)
  wmma_gemm_f32<4, 64, ACTA_GELU, ACTO_NONE, EPI_SKIPMIX>
      <<<gemm_grid(NP, 64, 4), blk, 0, stream>>>(
      aggw, 64, a_W_p, 64, 1, a_b_p, aggc, 64, xp, skip_p, NP);

  // 9) out = hp @ out_W + out_b
  out_kernel<<<dim3((NP + 255) / 256), blk, 0, stream>>>(aggc, out_W, out_b,
                                                         (float*)d_out, NP);
}